// NodeGCN_68856915690266
// MI455X (gfx1250) — compile-verified
//
#include <hip/hip_runtime.h>
#include <hip/hip_bf16.h>

typedef __attribute__((ext_vector_type(2))) float v2f;
typedef __attribute__((ext_vector_type(8))) float v8f;

#define HID 20

__device__ __forceinline__ void atomic_add_f32(float* p, float v) {
  (void)__hip_atomic_fetch_add(p, v, __ATOMIC_RELAXED, __HIP_MEMORY_SCOPE_AGENT);
}

// ---------------- degree / dinv ----------------
__global__ void deg_init_kernel(float* __restrict__ deg, int n) {
  int i = blockIdx.x * blockDim.x + threadIdx.x;
  if (i < n) deg[i] = 1.0f;   // self-loop weight 1
}

__global__ void deg_edge_kernel(const long long* __restrict__ dst,
                                const float* __restrict__ ew,
                                float* __restrict__ deg, int E) {
  int e = blockIdx.x * blockDim.x + threadIdx.x;
  if (e < E) atomic_add_f32(deg + (int)dst[e], ew[e]);
}

__global__ void deg_rsqrt_kernel(float* __restrict__ deg, int n) {
  int i = blockIdx.x * blockDim.x + threadIdx.x;
  if (i < n) deg[i] = rsqrtf(deg[i]);   // deg >= 1 always
}

// ---------------- WMMA GEMM: C[nrows,Fout] = A[nrows,K] @ W[K,Fout] ----------------
// One wave32 computes one 16x16 output tile with V_WMMA_F32_16X16X4_F32.
// K and FPAD (= tiles_n*16) are compile-time constants so the zero-padded
// LDS copy of W is addressed with immediate DS offsets (base + k*FPAD*4,
// max 16KB fits the 16-bit DS offset field) -- no per-iteration VALU
// address math. Requires nrows % 16 == 0 (100000 = 6250*16), K % 4 == 0.
template <int K, int FPAD>
__global__ void __launch_bounds__(256)
gemm_wmma_kernel(const float* __restrict__ A, int lda,
                 const float* __restrict__ W, int Fout,
                 float* __restrict__ outC, int ldo,
                 float* __restrict__ agg,
                 const float* __restrict__ dinv,
                 const float* __restrict__ bias,
                 int nrows) {
  constexpr int TILES_N = FPAD >> 4;
  __shared__ float wlds[K * FPAD];                // zero-padded weight tile

  for (int idx = threadIdx.x; idx < K * FPAD; idx += 256) {
    const int kk = idx / FPAD;
    const int cc = idx - kk * FPAD;
    wlds[idx] = (cc < Fout) ? W[kk * Fout + cc] : 0.0f;
  }
  __syncthreads();

  const int lane = threadIdx.x & 31;
  const int wave = (blockIdx.x * blockDim.x + threadIdx.x) >> 5;
  const int tiles_m = nrows >> 4;
  const int tm = wave / TILES_N;
  const int tn = wave - tm * TILES_N;
  if (tm >= tiles_m) return;            // wave-uniform exit: EXEC all-1s for WMMA

  const int half = lane & 15;           // M index (A) / N index (B)
  const int hi   = lane >> 4;           // selects K pair {0,1} vs {2,3}
  const int arow = tm * 16 + half;
  const int bcol = tn * 16 + half;      // bcol < FPAD always

  const float* __restrict__ Arow = A + (size_t)arow * lda + (hi << 1);
  const float* __restrict__ Bcol = wlds + (hi << 1) * FPAD + bcol;

  v8f c = {};
#pragma unroll
  for (int k0 = 0; k0 < K; k0 += 4) {
    // A 16x4: lanes 0-15 hold (M=half, K=k0..k0+1), lanes 16-31 hold K=k0+2..k0+3
    v2f a = *(const v2f*)(Arow + k0);
    // B 4x16: VGPR0 rows {k0, k0+2}, VGPR1 rows {k0+1, k0+3}, N striped over lanes
    v2f b;
    b.x = Bcol[k0 * FPAD];
    b.y = Bcol[k0 * FPAD + FPAD];
    c = __builtin_amdgcn_wmma_f32_16x16x4_f32(false, a, false, b, (short)0, c,
                                              false, false);
  }

  if (bcol < Fout) {
    const float bv = bias ? bias[bcol] : 0.0f;
#pragma unroll
    for (int r = 0; r < 8; ++r) {
      // C/D layout: VGPR r -> M = r (lanes 0-15) or r+8 (lanes 16-31), N = half
      const int row = tm * 16 + r + hi * 8;
      outC[(size_t)row * ldo + bcol] = c[r] + bv;
      if (agg) {
        const float di = dinv[row];
        agg[(size_t)row * HID + bcol] = di * di * c[r];
      }
    }
  }
}

// ---------------- edge aggregation: agg[dst] += dinv[s]*w*dinv[d] * xw[src] ----------------
__global__ void edge_agg_kernel(const long long* __restrict__ src,
                                const long long* __restrict__ dst,
                                const float* __restrict__ ew,
                                const float* __restrict__ dinv,
                                const float* __restrict__ xw,
                                float* __restrict__ agg, int E) {
  int e = blockIdx.x * blockDim.x + threadIdx.x;
  if (e >= E) return;
  const int s = (int)src[e];
  const int d = (int)dst[e];
  const float cf = dinv[s] * ew[e] * dinv[d];
  const float4* xs = (const float4*)(xw + (size_t)s * HID);  // 80B rows: 16B aligned
  float* ad = agg + (size_t)d * HID;
#pragma unroll
  for (int q = 0; q < 5; ++q) {
    float4 v = xs[q];
    atomic_add_f32(ad + q * 4 + 0, cf * v.x);
    atomic_add_f32(ad + q * 4 + 1, cf * v.y);
    atomic_add_f32(ad + q * 4 + 2, cf * v.z);
    atomic_add_f32(ad + q * 4 + 3, cf * v.w);
  }
}

// ---------------- bias + L2-normalize + ReLU, scatter into concat buffer ----------------
__global__ void post_kernel(const float* __restrict__ agg,
                            const float* __restrict__ bias,
                            float* __restrict__ cat, int off, int n) {
  int i = blockIdx.x * blockDim.x + threadIdx.x;
  if (i >= n) return;
  float v[HID];
  float ss = 0.0f;
  const float4* ag = (const float4*)(agg + (size_t)i * HID);
#pragma unroll
  for (int q = 0; q < 5; ++q) {
    float4 t = ag[q];
    v[q * 4 + 0] = t.x + bias[q * 4 + 0];
    v[q * 4 + 1] = t.y + bias[q * 4 + 1];
    v[q * 4 + 2] = t.z + bias[q * 4 + 2];
    v[q * 4 + 3] = t.w + bias[q * 4 + 3];
  }
#pragma unroll
  for (int f = 0; f < HID; ++f) ss += v[f] * v[f];
  const float inv = 1.0f / fmaxf(sqrtf(ss), 1e-12f);
  float* o = cat + (size_t)i * (3 * HID) + off;
#pragma unroll
  for (int f = 0; f < HID; ++f) o[f] = fmaxf(v[f] * inv, 0.0f);
}

// ---------------- launch ----------------
extern "C" void kernel_launch(void* const* d_in, const int* in_sizes, int n_in,
                              void* d_out, int out_size, void* d_ws, size_t ws_size,
                              hipStream_t stream) {
  const float*     x   = (const float*)d_in[0];
  const long long* ei  = (const long long*)d_in[1];   // [2, E] int64
  const float*     ew  = (const float*)d_in[2];
  const float*     W1  = (const float*)d_in[3];
  const float*     b1  = (const float*)d_in[4];
  const float*     W2  = (const float*)d_in[5];
  const float*     b2  = (const float*)d_in[6];
  const float*     W3  = (const float*)d_in[7];
  const float*     b3  = (const float*)d_in[8];
  const float*     Wl  = (const float*)d_in[9];
  const float*     bl  = (const float*)d_in[10];
  float*           out = (float*)d_out;

  const int FIN = 128;
  const int NC  = 10;
  const int n = in_sizes[0] / FIN;   // 100000 (multiple of 16)
  const int E = in_sizes[2];         // 3200000

  const long long* srcp = ei;
  const long long* dstp = ei + E;

  // workspace layout (256B aligned): dinv | xw | agg | cat  (~40.4 MB)
  char* ws = (char*)d_ws;
  auto al = [](size_t v) { return (v + 255) & ~(size_t)255; };
  size_t o = 0;
  float* dinv = (float*)(ws + o); o = al(o + (size_t)n * 4);
  float* xw   = (float*)(ws + o); o = al(o + (size_t)n * HID * 4);
  float* agg  = (float*)(ws + o); o = al(o + (size_t)n * HID * 4);
  float* cat  = (float*)(ws + o); o = al(o + (size_t)n * 3 * HID * 4);

  const int TB = 256;
  const int gN = (n + TB - 1) / TB;
  const int gE = (E + TB - 1) / TB;

  // dinv = rsqrt(segment_sum(ew, dst) + 1)
  deg_init_kernel<<<gN, TB, 0, stream>>>(dinv, n);
  deg_edge_kernel<<<gE, TB, 0, stream>>>(dstp, ew, dinv, E);
  deg_rsqrt_kernel<<<gN, TB, 0, stream>>>(dinv, n);

  const int tiles_m = n / 16;
  const int wavesL  = tiles_m * 2;                    // Fout=20 -> 2 N-tiles
  const int gGemmL  = (wavesL * 32 + TB - 1) / TB;

  // Layer 1: A = x [N,128]
  gemm_wmma_kernel<128, 32><<<gGemmL, TB, 0, stream>>>(
      x, FIN, W1, HID, xw, HID, agg, dinv, nullptr, n);
  edge_agg_kernel<<<gE, TB, 0, stream>>>(srcp, dstp, ew, dinv, xw, agg, E);
  post_kernel<<<gN, TB, 0, stream>>>(agg, b1, cat, 0, n);

  // Layer 2: A = out1 = cat[:, 0:20], lda = 60
  gemm_wmma_kernel<HID, 32><<<gGemmL, TB, 0, stream>>>(
      cat, 3 * HID, W2, HID, xw, HID, agg, dinv, nullptr, n);
  edge_agg_kernel<<<gE, TB, 0, stream>>>(srcp, dstp, ew, dinv, xw, agg, E);
  post_kernel<<<gN, TB, 0, stream>>>(agg, b2, cat, HID, n);

  // Layer 3: A = out2 = cat[:, 20:40]
  gemm_wmma_kernel<HID, 32><<<gGemmL, TB, 0, stream>>>(
      cat + HID, 3 * HID, W3, HID, xw, HID, agg, dinv, nullptr, n);
  edge_agg_kernel<<<gE, TB, 0, stream>>>(srcp, dstp, ew, dinv, xw, agg, E);
  post_kernel<<<gN, TB, 0, stream>>>(agg, b3, cat, 2 * HID, n);

  // Final linear: out[N,10] = cat[N,60] @ Wl + bl   (Fout=10 -> 1 N-tile)
  const int wavesF = tiles_m;
  const int gGemmF = (wavesF * 32 + TB - 1) / TB;
  gemm_wmma_kernel<3 * HID, 16><<<gGemmF, TB, 0, stream>>>(
      cat, 3 * HID, Wl, NC, out, NC, nullptr, nullptr, bl, n);
}